// BoxRoI_80822694576324
// MI455X (gfx1250) — compile-verified
//
#include <hip/hip_runtime.h>
#include <hip/hip_bf16.h>
#include <math.h>

#define NUM_CLASSES 81
#define NFG 80                      // foreground classes
#define N_PROP 2048
#define BATCH 4
#define PROB_TH 0.05f
#define NMS_TH 0.5f
#define DET_PER_IMG 100
#define MAX_OFF 4.135166556742356f  // log(1000/16)
#define OUT_COLS (NUM_CLASSES * 4 + NUM_CLASSES)   // 405

typedef unsigned int v4u __attribute__((ext_vector_type(4)));
typedef int          v8i __attribute__((ext_vector_type(8)));
typedef int          v4i __attribute__((ext_vector_type(4)));

// ---------------------------------------------------------------------------
// TDM: 1-D tensor load (contiguous nelem 4-byte elements) global -> LDS.
// D# group layout per CDNA5 ISA ch.8 (CDNA5 ISA):
//   group0: [1:0]=count, [63:32]=lds_addr, [120:64]=global_addr, [127:126]=type(2)
//   group1: [17:16]=data_size(2 => 4B), [79:48]=tensor_dim0, [127:112]=tile_dim0,
//           [143:128]=tile_dim1(=1), [207:160..]=tensor_dim0_stride
// groups 2/3 zero (<=2D tensor). workgroup_mask=0 (not in a cluster).
// This toolchain's builtin takes 6 args (g0, g1, g2, g3, g4, cpol).
// ---------------------------------------------------------------------------
__device__ __forceinline__ void tdm_load_1d(unsigned lds_off, const void* gptr,
                                            unsigned nelem) {
  unsigned long long ga = (unsigned long long)gptr;
  v4u g0;
  g0.x = 1u;                                              // count=1, user mode
  g0.y = lds_off;                                         // LDS byte address
  g0.z = (unsigned)(ga & 0xFFFFFFFFull);                  // global_addr[31:0]
  g0.w = (unsigned)((ga >> 32) & 0x1FFFFFFull) | (2u << 30); // [56:32] | type=2
  v8i g1;
  g1.s0 = (int)(2u << 16);                                // data_size = 4 bytes
  g1.s1 = (int)((nelem & 0xFFFFu) << 16);                 // tensor_dim0[15:0]
  g1.s2 = (int)((nelem >> 16) & 0xFFFFu);                 // tensor_dim0[31:16]
  g1.s3 = (int)((nelem & 0xFFFFu) << 16);                 // tile_dim0
  g1.s4 = 1;                                              // tile_dim1=1, tile_dim2=0
  g1.s5 = (int)nelem;                                     // tensor_dim0_stride lo
  g1.s6 = 0;
  g1.s7 = 0;
  v4i z4 = {0, 0, 0, 0};
  v8i z8 = {0, 0, 0, 0, 0, 0, 0, 0};
  __builtin_amdgcn_tensor_load_to_lds(g0, g1, z4, z4, z8, 0);
}

// ---------------------------------------------------------------------------
// Kernel A: decode + clip + softmax + threshold.
// One block per (b, n); threads 0..80 each own one class.
// Writes: out boxes columns [0,324), zeroes score columns [324,405),
//         class-major cls_boxes [B][80][N][4], cls_scores [B][80][N].
// ---------------------------------------------------------------------------
__global__ __launch_bounds__(96) void decode_softmax_kernel(
    const float* __restrict__ proposals, const float* __restrict__ regs,
    const float* __restrict__ logits, const float* __restrict__ sizes,
    float* __restrict__ out, float* __restrict__ cls_boxes,
    float* __restrict__ cls_scores) {
  const int bn = blockIdx.x;
  const int b = bn >> 11;
  const int n = bn & (N_PROP - 1);
  const int c = threadIdx.x;

  __shared__ float sl[NUM_CLASSES];
  __shared__ float s_red[2];  // [0]=max, [1]=sum

  const float* lrow = logits + (size_t)bn * NUM_CLASSES;
  if (c < NUM_CLASSES) sl[c] = lrow[c];
  __syncthreads();
  if (threadIdx.x == 0) {
    float m = -1e30f;
    for (int i = 0; i < NUM_CLASSES; ++i) m = fmaxf(m, sl[i]);
    s_red[0] = m;
  }
  __syncthreads();
  float e = 0.0f;
  if (c < NUM_CLASSES) { e = expf(sl[c] - s_red[0]); sl[c] = e; }
  __syncthreads();
  if (threadIdx.x == 0) {
    float s = 0.0f;
    for (int i = 0; i < NUM_CLASSES; ++i) s += sl[i];
    s_red[1] = s;
  }
  __syncthreads();

  const float4 p = *(const float4*)(proposals + (size_t)bn * 4);
  const float ws = p.z - p.x + 1.0f;
  const float hs = p.w - p.y + 1.0f;
  const float xc = p.x + 0.5f * ws;
  const float yc = p.y + 0.5f * hs;
  const float H = sizes[b * 2 + 0];
  const float W = sizes[b * 2 + 1];
  float* orow = out + (size_t)bn * OUT_COLS;

  if (c < NUM_CLASSES) {
    const float4 r = *(const float4*)(regs + (size_t)bn * (NUM_CLASSES * 4) + c * 4);
    const float dx = r.x * 0.1f;
    const float dy = r.y * 0.1f;
    const float dw = fminf(r.z * 0.2f, MAX_OFF);
    const float dh = fminf(r.w * 0.2f, MAX_OFF);
    const float xcc = xc + dx * ws;
    const float ycc = yc + dy * hs;
    const float w = ws * expf(dw);
    const float h = hs * expf(dh);
    float x1 = xcc - 0.5f * w;
    float y1 = ycc - 0.5f * h;
    float x2 = xcc + 0.5f * w - 1.0f;
    float y2 = ycc + 0.5f * h - 1.0f;
    x1 = fminf(fmaxf(x1, 0.0f), W - 1.0f);
    y1 = fminf(fmaxf(y1, 0.0f), H - 1.0f);
    x2 = fminf(fmaxf(x2, 0.0f), W - 1.0f);
    y2 = fminf(fmaxf(y2, 0.0f), H - 1.0f);

    orow[c * 4 + 0] = x1;
    orow[c * 4 + 1] = y1;
    orow[c * 4 + 2] = x2;
    orow[c * 4 + 3] = y2;
    orow[NUM_CLASSES * 4 + c] = 0.0f;     // score cols zeroed; topk fills later

    const float prob = e / s_red[1];
    const float score = (prob > PROB_TH) ? prob : 0.0f;
    if (c >= 1) {
      const size_t cb = (size_t)(b * NFG + (c - 1)) * N_PROP + n;
      float4 bx; bx.x = x1; bx.y = y1; bx.z = x2; bx.w = y2;
      *(float4*)(cls_boxes + cb * 4) = bx;
      cls_scores[cb] = score;
    }
  }
}

// ---------------------------------------------------------------------------
// Kernel B: per-(image,class) greedy NMS. One block per (b, class).
// Stages boxes + scores into LDS with TDM, bitonic-sorts (score desc, idx asc)
// keys in LDS, then the classic serial-rank / parallel-suppress greedy loop.
// Writes surviving scores back in place over cls_scores.
// Dynamic LDS layout (58 KB + eps):
//   [0,16384)      u64 keys[2048]
//   [16384,49152)  f32 boxes[2048*4]      (TDM target)
//   [49152,57344)  f32 sa[2048]           (TDM score target, then area)
//   [57344,59392)  u8  supp[2048]
//   [59392,59396)  int cnt
// ---------------------------------------------------------------------------
#define NMS_THREADS 1024

__global__ __launch_bounds__(NMS_THREADS) void nms_kernel(
    const float* __restrict__ cls_boxes, float* __restrict__ cls_scores) {
  extern __shared__ char smem[];
  unsigned long long* keys = (unsigned long long*)smem;
  float* boxes = (float*)(smem + 16384);
  float* sa    = (float*)(smem + 49152);
  unsigned char* supp = (unsigned char*)(smem + 57344);
  int* cntp = (int*)(smem + 59392);

  const int blk = blockIdx.x;                    // b * 80 + (class-1)
  const size_t base = (size_t)blk * N_PROP;
  const float* gboxes = cls_boxes + base * 4;
  float* gscores = cls_scores + base;
  const int tid = threadIdx.x;

  if (tid == 0) *cntp = 0;

  // Wave 0 issues the async tensor DMAs and waits on TENSORcnt; the
  // workgroup barrier then publishes the LDS tiles to all 32 waves.
  if (tid < 32) {
    const unsigned lds_boxes = (unsigned)(unsigned long long)(void*)boxes;
    const unsigned lds_sc    = (unsigned)(unsigned long long)(void*)sa;
    tdm_load_1d(lds_boxes, (const void*)gboxes, N_PROP * 4);
    tdm_load_1d(lds_sc, (const void*)gscores, N_PROP);
    __builtin_amdgcn_s_wait_tensorcnt(0);
  }
  __syncthreads();

  // Build sort keys: (score_bits << 32) | (2047 - n) so that a descending
  // u64 sort gives score-desc with stable idx-asc tie-break (matches
  // jnp.argsort(-scores)). Inactive boxes get key 0 and sink to the tail.
  int myact = 0;
  for (int n = tid; n < N_PROP; n += NMS_THREADS) {
    const float4 bx = ((const float4*)boxes)[n];
    const float ar = (bx.z - bx.x + 1.0f) * (bx.w - bx.y + 1.0f);
    const float sc = sa[n];
    if (sc > 0.0f) {
      keys[n] = ((unsigned long long)__float_as_uint(sc) << 32) |
                (unsigned)(N_PROP - 1 - n);
      ++myact;
    } else {
      keys[n] = 0ull;
    }
    sa[n] = ar;        // reuse the score buffer for per-box area
    supp[n] = 0;
  }
  if (myact) atomicAdd(cntp, myact);
  __syncthreads();
  const int K = *cntp;

  // Bitonic sort, descending, 2048 u64 keys in LDS.
  for (int k = 2; k <= N_PROP; k <<= 1) {
    for (int j = k >> 1; j > 0; j >>= 1) {
      for (int i = tid; i < N_PROP; i += NMS_THREADS) {
        const int ixj = i ^ j;
        if (ixj > i) {
          const unsigned long long a = keys[i];
          const unsigned long long bk = keys[ixj];
          const bool descSeg = ((i & k) == 0);
          const bool doswap = descSeg ? (a < bk) : (a > bk);
          if (doswap) { keys[i] = bk; keys[ixj] = a; }
        }
      }
      __syncthreads();
    }
  }

  // Greedy NMS: serial over ranks, parallel suppression of lower ranks.
  for (int r = 0; r < K; ++r) {
    if (!supp[r]) {                     // uniform: LDS value synced each iter
      const unsigned long long kr = keys[r];
      const int ir = N_PROP - 1 - (int)(unsigned)(kr & 0xFFFFFFFFull);
      const float4 br = ((const float4*)boxes)[ir];
      const float ar = sa[ir];
      for (int jj = r + 1 + tid; jj < K; jj += NMS_THREADS) {
        if (!supp[jj]) {
          const int ij = N_PROP - 1 - (int)(unsigned)(keys[jj] & 0xFFFFFFFFull);
          const float4 bj = ((const float4*)boxes)[ij];
          const float xtl = fmaxf(br.x, bj.x);
          const float ytl = fmaxf(br.y, bj.y);
          const float xbr = fminf(br.z, bj.z);
          const float ybr = fminf(br.w, bj.w);
          const float iw = fmaxf(xbr - xtl + 1.0f, 0.0f);
          const float ih = fmaxf(ybr - ytl + 1.0f, 0.0f);
          const float inter = iw * ih;
          const float iou = inter / (ar + sa[ij] - inter);
          if (iou > NMS_TH) supp[jj] = 1;
        }
      }
    }
    __syncthreads();
  }

  // Write back: zero slice, then scatter survivors (disjoint indices).
  for (int n = tid; n < N_PROP; n += NMS_THREADS) gscores[n] = 0.0f;
  __syncthreads();
  for (int r = tid; r < K; r += NMS_THREADS) {
    if (!supp[r]) {
      const unsigned long long kr = keys[r];
      const int ir = N_PROP - 1 - (int)(unsigned)(kr & 0xFFFFFFFFull);
      gscores[ir] = __uint_as_float((unsigned)(kr >> 32));
    }
  }
}

// ---------------------------------------------------------------------------
// Kernel C: per-image top-100 across 80*2048 surviving scores.
// Exact 100th value via binary search on positive-float bit patterns
// (monotone as uint); ties resolved by lowest flat index (matches lax.top_k
// stability). Scatters selected scores into the output score columns.
// ---------------------------------------------------------------------------
#define TOPK_THREADS 1024

__global__ __launch_bounds__(TOPK_THREADS) void topk_kernel(
    const float* __restrict__ cls_scores, float* __restrict__ out) {
  const int b = blockIdx.x;
  const float* kept = cls_scores + (size_t)b * NFG * N_PROP;
  const int M = NFG * N_PROP;
  const int tid = threadIdx.x;

  __shared__ int s_cnt;
  __shared__ int s_eq_cnt;
  __shared__ int s_eq_idx[1024];

  __builtin_prefetch(kept + tid * 64, 0, 0);   // global_prefetch_b8

  if (tid == 0) s_cnt = 0;
  __syncthreads();
  int my = 0;
  for (int f = tid; f < M; f += TOPK_THREADS) my += (kept[f] > 0.0f) ? 1 : 0;
  if (my) atomicAdd(&s_cnt, my);
  __syncthreads();
  const int P = s_cnt;

  unsigned Tbits = 0u;
  int G = P;
  if (P > DET_PER_IMG) {
    unsigned lo = 0u, hi = 0x40000000u;        // (0.0, 2.0]; scores <= 1.0
    while (hi - lo > 1u) {
      const unsigned mid = (lo + hi) >> 1;
      __syncthreads();
      if (tid == 0) s_cnt = 0;
      __syncthreads();
      int cc = 0;
      for (int f = tid; f < M; f += TOPK_THREADS)
        cc += (__float_as_uint(kept[f]) > mid) ? 1 : 0;
      if (cc) atomicAdd(&s_cnt, cc);
      __syncthreads();
      if (s_cnt >= DET_PER_IMG) lo = mid; else hi = mid;
    }
    Tbits = hi;                                 // bits of the 100th value
    __syncthreads();
    if (tid == 0) { s_cnt = 0; s_eq_cnt = 0; }
    __syncthreads();
    int cg = 0;
    for (int f = tid; f < M; f += TOPK_THREADS) {
      const unsigned vb = __float_as_uint(kept[f]);
      if (vb > Tbits) {
        ++cg;
      } else if (vb == Tbits) {
        const int slot = atomicAdd(&s_eq_cnt, 1);
        if (slot < 1024) s_eq_idx[slot] = f;
      }
    }
    if (cg) atomicAdd(&s_cnt, cg);
    __syncthreads();
    G = s_cnt;
  }

  // Scatter strictly-greater selections (or all positives if P <= 100).
  for (int f = tid; f < M; f += TOPK_THREADS) {
    const float v = kept[f];
    bool sel;
    if (P <= DET_PER_IMG) sel = (v > 0.0f);
    else                  sel = (__float_as_uint(v) > Tbits);
    if (sel) {
      const int cf = f >> 11;            // class-1
      const int n = f & (N_PROP - 1);
      out[((size_t)b * N_PROP + n) * OUT_COLS + NUM_CLASSES * 4 + (cf + 1)] = v;
    }
  }
  // Ties at the threshold: first (100 - G) by flat index.
  if (P > DET_PER_IMG) {
    __syncthreads();
    if (tid == 0) {
      int E = s_eq_cnt; if (E > 1024) E = 1024;
      for (int i = 1; i < E; ++i) {            // insertion sort ascending
        const int v = s_eq_idx[i];
        int j = i - 1;
        while (j >= 0 && s_eq_idx[j] > v) { s_eq_idx[j + 1] = s_eq_idx[j]; --j; }
        s_eq_idx[j + 1] = v;
      }
      const int slots = DET_PER_IMG - G;
      const float tv = __uint_as_float(Tbits);
      for (int i = 0; i < slots && i < E; ++i) {
        const int f = s_eq_idx[i];
        const int cf = f >> 11;
        const int n = f & (N_PROP - 1);
        out[((size_t)b * N_PROP + n) * OUT_COLS + NUM_CLASSES * 4 + (cf + 1)] = tv;
      }
    }
  }
}

// ---------------------------------------------------------------------------
extern "C" void kernel_launch(void* const* d_in, const int* in_sizes, int n_in,
                              void* d_out, int out_size, void* d_ws, size_t ws_size,
                              hipStream_t stream) {
  (void)in_sizes; (void)n_in; (void)out_size; (void)ws_size;
  const float* proposals = (const float*)d_in[0];   // [B, N, 4]
  const float* regs      = (const float*)d_in[1];   // [B, N, 81*4]
  const float* logits    = (const float*)d_in[2];   // [B, N, 81]
  const float* sizes     = (const float*)d_in[3];   // [B, 2]
  float* out = (float*)d_out;                       // [B, N, 405]

  float* cls_boxes  = (float*)d_ws;                                   // B*80*N*4
  float* cls_scores = cls_boxes + (size_t)BATCH * NFG * N_PROP * 4;   // B*80*N

  decode_softmax_kernel<<<BATCH * N_PROP, 96, 0, stream>>>(
      proposals, regs, logits, sizes, out, cls_boxes, cls_scores);

  const size_t smem = 16384 + 32768 + 8192 + 2048 + 64;   // 59456 B < 64 KB
  nms_kernel<<<BATCH * NFG, NMS_THREADS, smem, stream>>>(cls_boxes, cls_scores);

  topk_kernel<<<BATCH, TOPK_THREADS, 0, stream>>>(cls_scores, out);
}